// MaxMarginLoss_8375186227913
// MI455X (gfx1250) — compile-verified
//
#include <hip/hip_runtime.h>
#include <hip/hip_bf16.h>

typedef __attribute__((ext_vector_type(2))) float v2f;
typedef __attribute__((ext_vector_type(8))) float v8f;

#define NB 32
#define NT 2048
#define ND 1024
#define NS 32
#define ALPHA_C 1.0f

// ---------------- Kernel 1: per-(b,step) counts and first-occurrence position
__global__ void k_counts(const int* __restrict__ step_ids,
                         float* __restrict__ counts, int* __restrict__ pos) {
  int b = blockIdx.x;
  __shared__ int c[NS];
  __shared__ int p[NS];
  int tid = threadIdx.x;
  if (tid < NS) { c[tid] = 0; p[tid] = NT; }
  __syncthreads();
  const int* sid = step_ids + b * NT;
  for (int t = tid; t < NT; t += blockDim.x) {
    int s = sid[t];
    if (s >= 1 && s <= NS) {
      atomicAdd(&c[s - 1], 1);
      atomicMin(&p[s - 1], t);
    }
  }
  __syncthreads();
  if (tid < NS) {
    counts[b * NS + tid] = (float)c[tid];
    pos[b * NS + tid] = p[tid];
  }
}

// ---------------- Kernel 2: h[b,s,d] = (mask^T @ |x|) / max(count,1) via f32 WMMA
// grid.x = NB * 8 ; block = 256 (8 waves); each wave owns 16 d-columns and both
// 16-step tiles. A-fragment (mask) built from LDS step_ids; B-fragment = |x|.
__global__ void k_einsum_wmma(const float* __restrict__ x,
                              const int* __restrict__ step_ids,
                              const float* __restrict__ counts,
                              float* __restrict__ h) {
  int b = blockIdx.x >> 3;
  int tile = blockIdx.x & 7;

  __shared__ __align__(8) int s_sid[NT];
  const int* sid = step_ids + b * NT;
  for (int t = threadIdx.x; t < NT; t += blockDim.x) s_sid[t] = sid[t];
  __syncthreads();

  int wave = threadIdx.x >> 5;
  int lane = threadIdx.x & 31;
  int l16  = lane & 15;   // M index (step) for A, N index (col) for B/C
  int khi  = lane >> 4;   // selects K pair {0,1} vs {2,3}
  int col  = tile * 128 + wave * 16 + l16;

  const float* xb = x + (size_t)b * NT * ND + col;

  v8f c0 = {};  // steps 1..16
  v8f c1 = {};  // steps 17..32
  int step_lo = l16 + 1;
  int step_hi = l16 + 17;

#pragma unroll 4
  for (int t0 = 0; t0 < NT; t0 += 4) {
    int ta = t0 + 2 * khi;
    int2 sp = *(const int2*)&s_sid[ta];   // 8B-aligned (ta even)
    v2f a0, a1, bb;
    a0.x = (sp.x == step_lo) ? 1.0f : 0.0f;
    a0.y = (sp.y == step_lo) ? 1.0f : 0.0f;
    a1.x = (sp.x == step_hi) ? 1.0f : 0.0f;
    a1.y = (sp.y == step_hi) ? 1.0f : 0.0f;
    bb.x = __builtin_fabsf(xb[(size_t)ta * ND]);
    bb.y = __builtin_fabsf(xb[(size_t)(ta + 1) * ND]);
    c0 = __builtin_amdgcn_wmma_f32_16x16x4_f32(false, a0, false, bb,
                                               (short)0, c0, false, false);
    c1 = __builtin_amdgcn_wmma_f32_16x16x4_f32(false, a1, false, bb,
                                               (short)0, c1, false, false);
  }

  float* hb = h + (size_t)b * NS * ND;
#pragma unroll
  for (int r = 0; r < 8; ++r) {
    int s0 = r + 8 * khi;          // step index 0..15 (c0 tile)
    int s1 = s0 + 16;              // step index 16..31 (c1 tile)
    float n0 = fmaxf(counts[b * NS + s0], 1.0f);
    float n1 = fmaxf(counts[b * NS + s1], 1.0f);
    hb[(size_t)s0 * ND + col] = c0[r] / n0;
    hb[(size_t)s1 * ND + col] = c1[r] / n1;
  }
}

// ---------------- Kernel 3: per-batch ordering, pair energies, partial losses
__global__ void k_pairs(const float* __restrict__ h, const int* __restrict__ pos,
                        const int* __restrict__ labels,
                        float* __restrict__ outc, float* __restrict__ outn) {
  int b = blockIdx.x;
  __shared__ int s_pos[NS];
  __shared__ int s_ord[NS];
  __shared__ float s_E[NS - 1];
  __shared__ float red[256];
  int tid = threadIdx.x;

  if (tid < NS) s_pos[tid] = pos[b * NS + tid];
  __syncthreads();
  if (tid < NS) {  // stable rank-sort by pos (ties broken by index)
    int p = s_pos[tid];
    int r = 0;
    for (int j = 0; j < NS; ++j) {
      int q = s_pos[j];
      r += ((q < p) || (q == p && j < tid)) ? 1 : 0;
    }
    s_ord[r] = tid;
  }
  __syncthreads();

  const float* hb = h + (size_t)b * NS * ND;
  for (int k = 0; k < NS - 1; ++k) {
    int i = s_ord[k], j = s_ord[k + 1];
    float acc = 0.f;
    for (int d = tid; d < ND; d += 256) {
      float diff = hb[(size_t)i * ND + d] - hb[(size_t)j * ND + d];
      diff = fmaxf(diff, 0.f);
      acc += diff * diff;
    }
    red[tid] = acc;
    __syncthreads();
    for (int s = 128; s > 0; s >>= 1) {
      if (tid < s) red[tid] += red[tid + s];
      __syncthreads();
    }
    if (tid == 0) s_E[k] = red[0] * (1.0f / (float)ND);
    __syncthreads();
  }

  if (tid == 0) {
    int n = 0;
    for (int i = 0; i < NS; ++i) n += (s_pos[i] < NT) ? 1 : 0;
    float lp = 0.f, ln = 0.f;
    int np = 0, ni = 0;
    for (int k = 0; k < NS - 1; ++k) {
      bool pv = s_pos[s_ord[k + 1]] < NT;
      if (pv) {
        np++;
        lp += s_E[k];
        if (s_ord[k] > s_ord[k + 1]) {  // step values = index+1, same ordering
          ni++;
          ln += fmaxf(ALPHA_C - s_E[k], 0.f);
        }
      }
    }
    lp /= fmaxf((float)np, 1.f);
    ln /= fmaxf((float)ni, 1.f);
    int label = labels[b];
    bool pc = (label == 1) && (n >= 2);
    bool nc = (label == 0) && (ni > 0);
    outc[b] = (pc ? lp : 0.f) + (nc ? ln : 0.f);
    outn[b] = (pc ? 1.f : 0.f) + (nc ? 1.f : 0.f);
  }
}

// ---------------- Kernel 4: final scalar
__global__ void k_final(const float* __restrict__ c, const float* __restrict__ n,
                        float* __restrict__ out) {
  if (threadIdx.x == 0) {
    float tot = 0.f, num = 0.f;
    for (int b = 0; b < NB; ++b) { tot += c[b]; num += n[b]; }
    out[0] = tot / (num + 1e-9f);
  }
}

extern "C" void kernel_launch(void* const* d_in, const int* in_sizes, int n_in,
                              void* d_out, int out_size, void* d_ws, size_t ws_size,
                              hipStream_t stream) {
  const float* x      = (const float*)d_in[0];  // (32,2048,1024) f32
  const int*  sids    = (const int*)d_in[1];    // (32,2048) i32
  const int*  labels  = (const int*)d_in[2];    // (32,) i32
  float* out = (float*)d_out;

  char* ws = (char*)d_ws;
  float* h      = (float*)(ws);                                   // 4 MB
  float* counts = (float*)(ws + (size_t)NB * NS * ND * 4);        // 4 KB
  int*   pos    = (int*)  (ws + (size_t)NB * NS * ND * 4 + 4096); // 4 KB
  float* outc   = (float*)(ws + (size_t)NB * NS * ND * 4 + 8192);
  float* outn   = outc + NB;

  k_counts<<<NB, 256, 0, stream>>>(sids, counts, pos);
  k_einsum_wmma<<<NB * 8, 256, 0, stream>>>(x, sids, counts, h);
  k_pairs<<<NB, 256, 0, stream>>>(h, pos, labels, outc, outn);
  k_final<<<1, 32, 0, stream>>>(outc, outn, out);
}